// EdgeFeatInit_19542101197172
// MI455X (gfx1250) — compile-verified
//
#include <hip/hip_runtime.h>

// ---- problem constants (from reference) ----
#define EDGE_F 64
#define ATOM_F 128
#define HIDDEN 128
#define KTOT   (EDGE_F + ATOM_F)   // 192
#define TM     128                 // edges per workgroup
#define LDS_STRIDE 196             // padded row stride in floats: even (8B-aligned b64), 196%64==4 (bank-friendly)

typedef __attribute__((ext_vector_type(2))) float v2f;
typedef __attribute__((ext_vector_type(4))) float v4f;
typedef __attribute__((ext_vector_type(8))) float v8f;

// Async global -> LDS copy of 16 bytes per enabled lane (CDNA5, tracked by ASYNCcnt).
// lds_off: byte offset in the workgroup's LDS allocation (low 32 bits of a generic LDS pointer).
// gaddr:   per-lane 64-bit global address (16B aligned).
__device__ __forceinline__ void async_load_b128(unsigned lds_off, const void* gaddr) {
    asm volatile("global_load_async_to_lds_b128 %0, %1, off"
                 :: "v"(lds_off), "v"(gaddr)
                 : "memory");
}
__device__ __forceinline__ void wait_async_zero() {
    asm volatile("s_wait_asynccnt 0x0" ::: "memory");
}

__launch_bounds__(256, 1)
__global__ void edge_feat_init_wmma(const float* __restrict__ e_input,
                                    const float* __restrict__ n_input,
                                    const int*   __restrict__ src_idx,
                                    const float* __restrict__ W,
                                    float* __restrict__ out,
                                    int n_edges)
{
    // Single shared array so LDS offsets are contiguous/known for the async path.
    __shared__ __align__(16) float smem[TM * LDS_STRIDE + HIDDEN * LDS_STRIDE + TM];
    float* sFeat = smem;                                   // [TM][LDS_STRIDE]   concat features, K-contiguous
    float* sWt   = smem + TM * LDS_STRIDE;                 // [HIDDEN][LDS_STRIDE] W transposed, K-contiguous
    int*   sIdx  = (int*)(smem + TM * LDS_STRIDE + HIDDEN * LDS_STRIDE);

    const int tid      = threadIdx.x;
    const int lane     = tid & 31;
    const int wave     = tid >> 5;                 // 0..7
    const int edgeBase = blockIdx.x * TM;
    if (edgeBase >= n_edges) return;

    // ---- stage gather indices (normal path) ----
    if (tid < TM) sIdx[tid] = src_idx[edgeBase + tid];

    // ---- stage W transposed: sWt[n][k] = W[k*HIDDEN + n] (coalesced read; transpose => VGPR path) ----
    for (int i = tid; i < KTOT * HIDDEN; i += 256) {
        const int k = i / HIDDEN;
        const int n = i - k * HIDDEN;
        sWt[n * LDS_STRIDE + k] = W[i];
    }
    __syncthreads();   // sIdx visible before async gather addresses are formed

    // ---- async-stage concat features straight into LDS (CDNA5 GLOBAL_LOAD_ASYNC_TO_LDS_B128) ----
    // sFeat[row][0:64] = e_input[edgeBase+row], sFeat[row][64:192] = n_input[src_idx[row]]
    // 128 rows x 48 float4 = 6144 chunks; 24 per thread, per-lane addressing (gather-capable).
    for (int i = tid; i < TM * (KTOT / 4); i += 256) {
        const int row = i / (KTOT / 4);
        const int c4  = i - row * (KTOT / 4);
        const int col = c4 * 4;
        const float* g;
        if (col < EDGE_F) {
            g = e_input + (long)(edgeBase + row) * EDGE_F + col;
        } else {
            g = n_input + (long)sIdx[row] * ATOM_F + (col - EDGE_F);
        }
        const unsigned ldsOff = (unsigned)(uintptr_t)(sFeat + row * LDS_STRIDE + col);
        async_load_b128(ldsOff, g);
    }
    wait_async_zero();   // this wave's async copies landed in LDS
    __syncthreads();     // all waves' copies visible

    // ---- WMMA: each wave computes a 16-edge strip x all 128 outputs ----
    // A-frag (16x4 f32, ISA 7.12.2): lane l holds row M=(l&15), K = {2*(l>>4), 2*(l>>4)+1} in 2 VGPRs.
    // B-frag mirrored: lane l holds col N=(l&15), same K pair.
    const int mBase = wave * 16;
    const int rowA  = mBase + (lane & 15);
    const int kHalf = (lane >> 4) * 2;

    const float* aPtr = &sFeat[rowA * LDS_STRIDE + kHalf];
    const float* bPtr = &sWt[(lane & 15) * LDS_STRIDE + kHalf];

    v8f zero = {};
    v8f acc[8];
#pragma unroll
    for (int j = 0; j < 8; ++j) acc[j] = zero;

#pragma unroll 4
    for (int kk = 0; kk < KTOT; kk += 4) {
        const v2f aFrag = *(const v2f*)(aPtr + kk);
#pragma unroll
        for (int j = 0; j < 8; ++j) {
            const v2f bFrag = *(const v2f*)(bPtr + j * 16 * LDS_STRIDE + kk);
            acc[j] = __builtin_amdgcn_wmma_f32_16x16x4_f32(
                /*neg_a=*/false, aFrag,
                /*neg_b=*/false, bFrag,
                /*c_mod=*/(short)0, acc[j],
                /*reuse_a=*/false, /*reuse_b=*/false);
        }
    }

    // ---- store D tiles non-temporally (out is write-once; keep L2 for the gather table) ----
    // VGPR r: lanes 0-15 -> M=r, lanes 16-31 -> M=8+r; N = (lane&15)
    const int nCol = lane & 15;
    const int mOff = (lane >> 4) * 8;
#pragma unroll
    for (int j = 0; j < 8; ++j) {
        float* o = out + (long)(edgeBase + mBase + mOff) * HIDDEN + j * 16 + nCol;
#pragma unroll
        for (int r = 0; r < 8; ++r) {
            __builtin_nontemporal_store(acc[j][r], o + (long)r * HIDDEN);
        }
    }
}

extern "C" void kernel_launch(void* const* d_in, const int* in_sizes, int n_in,
                              void* d_out, int out_size, void* d_ws, size_t ws_size,
                              hipStream_t stream) {
    const float* e_input = (const float*)d_in[0];
    const float* n_input = (const float*)d_in[1];
    const int*   src_idx = (const int*)d_in[2];
    const float* W       = (const float*)d_in[3];
    float*       out     = (float*)d_out;

    const int n_edges = in_sizes[0] / EDGE_F;
    const int grid    = (n_edges + TM - 1) / TM;

    edge_feat_init_wmma<<<grid, 256, 0, stream>>>(e_input, n_input, src_idx, W, out, n_edges);
}